// HierarchicalDiracModel_16647293239906
// MI455X (gfx1250) — compile-verified
//
#include <hip/hip_runtime.h>
#include <hip/hip_bf16.h>

// out[n] = prod_{b=0..24} chosen[24-b][bit_b(n)]  (Kronecker product of 25
// softmax pairs). Factor n = hi*8192 + lo  ->  out = HI (4096) outer LO (8192).
// Outer product tiles computed on the matrix pipe via V_WMMA_F32_16X16X4_F32.

typedef __attribute__((ext_vector_type(2))) float v2f;
typedef __attribute__((ext_vector_type(8))) float v8f;

#define LOG2N   25
#define LO_BITS 13
#define HI_BITS 12
#define LO_SIZE (1 << LO_BITS)   // 8192 columns
#define HI_SIZE (1 << HI_BITS)   // 4096 rows

// ---------------------------------------------------------------------------
// Setup: softmax the 25 rows, build HI[] / LO[] factor tables in workspace.
// One block; negligible cost vs. 128 MB of streaming stores.
// ---------------------------------------------------------------------------
__global__ void dirac_setup(const float* __restrict__ elements,
                            float* __restrict__ hi_tab,
                            float* __restrict__ lo_tab) {
    __shared__ float c[LOG2N][2];
    const int t = threadIdx.x;
    if (t < LOG2N) {
        float e0 = elements[2 * t + 0];
        float e1 = elements[2 * t + 1];
        float m  = fmaxf(e0, e1);
        float x0 = __expf(e0 - m);
        float x1 = __expf(e1 - m);
        float inv = 1.0f / (x0 + x1);
        c[t][0] = x0 * inv;
        c[t][1] = x1 * inv;
    }
    __syncthreads();

    // LO[lo] = prod_{b=0..12} chosen[24-b][bit_b(lo)]
    for (int i = t; i < LO_SIZE; i += blockDim.x) {
        float p = 1.0f;
#pragma unroll
        for (int b = 0; b < LO_BITS; ++b) p *= c[LOG2N - 1 - b][(i >> b) & 1];
        lo_tab[i] = p;
    }
    // HI[hi] = prod_{k=0..11} chosen[11-k][bit_k(hi)]
    for (int i = t; i < HI_SIZE; i += blockDim.x) {
        float p = 1.0f;
#pragma unroll
        for (int k = 0; k < HI_BITS; ++k) p *= c[HI_BITS - 1 - k][(i >> k) & 1];
        hi_tab[i] = p;
    }
}

// ---------------------------------------------------------------------------
// Main: rank-1 update per 16x16 tile on the WMMA pipe, stream tiles to HBM.
// Output viewed as [4096][8192] f32 row-major.
// Wave -> one 16-row strip, 16 column tiles (16 KB written per wave).
// All loads are unconditional (lanes 16-31 re-read lane&15 addresses, same
// cachelines) and fragments are zero-masked with selects, so EXEC stays
// all-ones throughout -- no divergent-load branches, loads clause/pipeline
// ahead of the WMMA + store stream.
// ---------------------------------------------------------------------------
#define WAVES_PER_BLOCK 8
#define WAVES_PER_ROW   32                      // waves sharing one row strip
#define COL_TILES       (LO_SIZE / 16)          // 512
#define TILES_PER_WAVE  (COL_TILES / WAVES_PER_ROW)   // 16
#define ROW_TILES       (HI_SIZE / 16)          // 256
#define TOTAL_WAVES     (ROW_TILES * WAVES_PER_ROW)   // 8192
#define GRID_BLOCKS     (TOTAL_WAVES / WAVES_PER_BLOCK) // 1024

__global__ void __launch_bounds__(256)
dirac_wmma(const float* __restrict__ hi_tab,
           const float* __restrict__ lo_tab,
           float* __restrict__ out) {
    const int lane   = threadIdx.x & 31;
    const int wave   = blockIdx.x * WAVES_PER_BLOCK + (threadIdx.x >> 5);
    const int tileRow = wave / WAVES_PER_ROW;   // 0..255
    const int chunk   = wave % WAVES_PER_ROW;   // 0..31
    const int rowBase = tileRow * 16;
    const int lane16  = lane & 15;
    const bool lowHalf = (lane < 16);

    const int colStart = chunk * (TILES_PER_WAVE * 16);   // 256 cols per wave

    // Prefetch this wave's 1 KB LO chunk (global_prefetch_b8).
    __builtin_prefetch(lo_tab + colStart + lane, 0, 3);

    // A fragment (16x4 f32): only K=0 column nonzero -> lanes 0-15, VGPR0.
    // Unconditional load + mask (no EXEC manipulation).
    const float hv = hi_tab[rowBase + lane16];
    v2f a;
    a.x = lowHalf ? hv : 0.0f;
    a.y = 0.0f;

    // D layout: VGPR r, lanes 0-15 -> row rowBase+r, cols N=lane;
    //           lanes 16-31 -> row rowBase+r+8, cols N=lane-16.
    float* outp = out + (size_t)(rowBase + (lowHalf ? 0 : 8)) * LO_SIZE
                      + lane16 + colStart;
    const float* lop = lo_tab + colStart + lane16;

#pragma unroll
    for (int j = 0; j < TILES_PER_WAVE; ++j) {
        // B fragment (4x16 f32): only K=0 row nonzero -> lanes 0-15, VGPR0.
        const float lv = lop[j * 16];           // unconditional, cache-hit
        v2f b;
        b.x = lowHalf ? lv : 0.0f;
        b.y = 0.0f;

        v8f cz = {0.f, 0.f, 0.f, 0.f, 0.f, 0.f, 0.f, 0.f};
        // D = A x B + 0 : 16x16 outer-product tile on the matrix pipe.
        v8f d = __builtin_amdgcn_wmma_f32_16x16x4_f32(
            /*neg_a=*/false, a, /*neg_b=*/false, b,
            /*c_mod=*/(short)0, cz, /*reuse_a=*/false, /*reuse_b=*/false);

        float* p = outp + j * 16;
#pragma unroll
        for (int r = 0; r < 8; ++r) p[(size_t)r * LO_SIZE] = d[r];
    }
}

// ---------------------------------------------------------------------------
extern "C" void kernel_launch(void* const* d_in, const int* in_sizes, int n_in,
                              void* d_out, int out_size, void* d_ws, size_t ws_size,
                              hipStream_t stream) {
    const float* elements = (const float*)d_in[0];   // [25, 2] f32
    float* hi_tab = (float*)d_ws;                    // 4096 f32
    float* lo_tab = hi_tab + HI_SIZE;                // 8192 f32 (48 KB total)
    float* out    = (float*)d_out;                   // 2^25 f32

    dirac_setup<<<1, 256, 0, stream>>>(elements, hi_tab, lo_tab);
    dirac_wmma<<<GRID_BLOCKS, 256, 0, stream>>>(hi_tab, lo_tab, out);
}